// VecInt_60936995995696
// MI455X (gfx1250) — compile-verified
//
#include <hip/hip_runtime.h>

// VecInt: scaling-and-squaring integration of a stationary velocity field.
// vec: [B=2, C=3, D=160, H=192, W=160] fp32.
// vec *= 2^-7; repeat 7x: vec = vec + trilinear_warp(vec, vec) (zero padding).
//
// Memory-bound gather: ~2.5 GB total traffic -> ~107 us floor at 23.3 TB/s;
// one 118 MB tensor fits the 192 MB L2 so the 8-corner gathers are largely
// L2-resident. No contraction -> WMMA not applicable. CDNA5 levers used:
// wave32 full-EXEC launch geometry (no div/mod prologue), coalesced B32
// vmem with W innermost, global_prefetch_b8 for the ch1/ch2 gather lines.

#define D_ 160
#define H_ 192
#define W_ 160
#define B_ 2

static constexpr int HW_ = H_ * W_;       // 30720
static constexpr int S_  = D_ * H_ * W_;  // 4,915,200 (channel stride)
static constexpr int BS_ = 3 * S_;        // 14,745,600 (batch stride)

// Launch geometry supplies coordinates directly (no integer division):
//   threadIdx.x = w (0..159, 5 wave32s), blockIdx.x = h, blockIdx.y = d,
//   blockIdx.z = b. Every thread in-range -> EXEC all-ones throughout.
template <bool SCALED>
__global__ __launch_bounds__(W_)
void warp_step(const float* __restrict__ src, float* __restrict__ dst, float scale) {
    const int w = threadIdx.x;
    const int h = blockIdx.x;
    const int d = blockIdx.y;
    const int b = blockIdx.z;

    const float* p0 = src + b * BS_;          // channel 0 of this batch
    const int voff = (d * H_ + h) * W_ + w;

    // Flow values = identity-add terms (flow IS the source field).
    float v0 = p0[voff];
    float v1 = p0[S_ + voff];
    float v2 = p0[2 * S_ + voff];
    if (SCALED) { v0 *= scale; v1 *= scale; v2 *= scale; }

    // Sample coordinates (voxel units), spatial dim i displaced by channel i.
    float gd = (float)d + v0;
    float gh = (float)h + v1;
    float gw = (float)w + v2;

    float fd = floorf(gd), fh = floorf(gh), fw = floorf(gw);
    int d0 = (int)fd, h0 = (int)fh, w0 = (int)fw;
    int d1 = d0 + 1,  h1 = h0 + 1,  w1 = w0 + 1;
    float td = gd - fd, th = gh - fh, tw = gw - fw;

    // Branchless zero-padding (map_coordinates order=1, mode='constant',
    // cval=0): zero the per-axis weight of any out-of-range corner, clamp
    // its index so the load stays in-bounds.
    float ad0 = (1.0f - td) * (((unsigned)d0 < (unsigned)D_) ? 1.0f : 0.0f);
    float ad1 =         td  * (((unsigned)d1 < (unsigned)D_) ? 1.0f : 0.0f);
    float ah0 = (1.0f - th) * (((unsigned)h0 < (unsigned)H_) ? 1.0f : 0.0f);
    float ah1 =         th  * (((unsigned)h1 < (unsigned)H_) ? 1.0f : 0.0f);
    float aw0 = (1.0f - tw) * (((unsigned)w0 < (unsigned)W_) ? 1.0f : 0.0f);
    float aw1 =         tw  * (((unsigned)w1 < (unsigned)W_) ? 1.0f : 0.0f);

    int cd0 = min(max(d0, 0), D_ - 1), cd1 = min(max(d1, 0), D_ - 1);
    int ch0 = min(max(h0, 0), H_ - 1), ch1 = min(max(h1, 0), H_ - 1);
    int cw0 = min(max(w0, 0), W_ - 1), cw1 = min(max(w1, 0), W_ - 1);

    int od0 = cd0 * HW_, od1 = cd1 * HW_;
    int oh0 = ch0 * W_,  oh1 = ch1 * W_;

    int o000 = od0 + oh0 + cw0, o001 = od0 + oh0 + cw1;
    int o010 = od0 + oh1 + cw0, o011 = od0 + oh1 + cw1;
    int o100 = od1 + oh0 + cw0, o101 = od1 + oh0 + cw1;
    int o110 = od1 + oh1 + cw0, o111 = od1 + oh1 + cw1;

    // Warm cache for channel 1/2 gather lines while channel-0 math runs
    // (lowers to global_prefetch_b8 on gfx1250).
    __builtin_prefetch(p0 + S_     + o000, 0, 3);
    __builtin_prefetch(p0 + 2 * S_ + o000, 0, 3);

    float wt000 = ad0 * ah0 * aw0, wt001 = ad0 * ah0 * aw1;
    float wt010 = ad0 * ah1 * aw0, wt011 = ad0 * ah1 * aw1;
    float wt100 = ad1 * ah0 * aw0, wt101 = ad1 * ah0 * aw1;
    float wt110 = ad1 * ah1 * aw0, wt111 = ad1 * ah1 * aw1;

    float id[3] = {v0, v1, v2};
    float* q = dst + b * BS_ + voff;

#pragma unroll
    for (int c = 0; c < 3; ++c) {
        const float* p = p0 + c * S_;
        float acc;
        acc = wt000 * p[o000];
        acc = fmaf(wt001, p[o001], acc);
        acc = fmaf(wt010, p[o010], acc);
        acc = fmaf(wt011, p[o011], acc);
        acc = fmaf(wt100, p[o100], acc);
        acc = fmaf(wt101, p[o101], acc);
        acc = fmaf(wt110, p[o110], acc);
        acc = fmaf(wt111, p[o111], acc);
        // SCALED: gathered raw values also carry the 2^-7 factor.
        q[c * S_] = SCALED ? fmaf(scale, acc, id[c]) : (id[c] + acc);
    }
}

extern "C" void kernel_launch(void* const* d_in, const int* in_sizes, int n_in,
                              void* d_out, int out_size, void* d_ws, size_t ws_size,
                              hipStream_t stream) {
    const float* vec = (const float*)d_in[0];
    float* out = (float*)d_out;
    float* ws  = (float*)d_ws;   // needs >= 117,964,800 bytes (one tensor)

    const dim3 blk(W_);                 // 160 threads = 5 wave32s
    const dim3 grd(H_, D_, B_);         // 192 x 160 x 2 = 61,440 blocks
    const float scale = 1.0f / 128.0f;  // 2^-NSTEPS, NSTEPS = 7

    // Step 1 fuses the 2^-7 prescale (reads d_in, never written).
    // Parity chosen so step 7 lands in d_out:
    //   in->out, out->ws, ws->out, out->ws, ws->out, out->ws, ws->out
    warp_step<true ><<<grd, blk, 0, stream>>>(vec, out, scale);
    warp_step<false><<<grd, blk, 0, stream>>>(out, ws,  1.0f);
    warp_step<false><<<grd, blk, 0, stream>>>(ws,  out, 1.0f);
    warp_step<false><<<grd, blk, 0, stream>>>(out, ws,  1.0f);
    warp_step<false><<<grd, blk, 0, stream>>>(ws,  out, 1.0f);
    warp_step<false><<<grd, blk, 0, stream>>>(out, ws,  1.0f);
    warp_step<false><<<grd, blk, 0, stream>>>(ws,  out, 1.0f);
    (void)in_sizes; (void)n_in; (void)out_size; (void)ws_size;
}